// GINLayer_59287728554193
// MI455X (gfx1250) — compile-verified
//
#include <hip/hip_runtime.h>

typedef __attribute__((ext_vector_type(2))) float v2f;
typedef __attribute__((ext_vector_type(8))) float v8f;

#define LDS_STRIDE 68   // 68 floats = 272 bytes: 16B-aligned rows, conflict-free A-frag reads

// ---------------- kernel 1: zero the aggregation scratch ----------------
__global__ void gin_zero_kernel(float4* __restrict__ p, int n4) {
  int i = blockIdx.x * blockDim.x + threadIdx.x;
  if (i < n4) p[i] = make_float4(0.f, 0.f, 0.f, 0.f);
}

// ---------------- kernel 2: edge scatter-add (segment_sum) ----------------
// One wave32 per edge: lane L handles channels 2L, 2L+1.
__global__ void gin_scatter_kernel(const float* __restrict__ x,
                                   const int* __restrict__ ei,
                                   float* __restrict__ agg,
                                   int n_edges) {
  int gtid = blockIdx.x * blockDim.x + threadIdx.x;
  int edge = gtid >> 5;
  int lane = threadIdx.x & 31;
  if (edge >= n_edges) return;
  int src = ei[edge];            // edge_index[0][e]
  int dst = ei[n_edges + edge];  // edge_index[1][e]
  const float2 v = *(const float2*)(x + (size_t)src * 64 + lane * 2);
  float* p = agg + (size_t)dst * 64 + lane * 2;
  atomicAdd(p + 0, v.x);
  atomicAdd(p + 1, v.y);
}

// ---------------- kernel 3: fused (1+eps)x + agg -> Linear+ReLU -> Linear ----------------
// One wave per 16-node tile; fp32 WMMA 16x16x4, M=16 N=64 K=64 per layer.
__global__ __launch_bounds__(256) void gin_mlp_kernel(
    const float* __restrict__ x, const float* __restrict__ agg,
    const float* __restrict__ eps_p,
    const float* __restrict__ W1, const float* __restrict__ b1,
    const float* __restrict__ W2, const float* __restrict__ b2,
    float* __restrict__ out, int n_nodes) {
  __shared__ float lds[8 * 16 * LDS_STRIDE];

  const int tid  = threadIdx.x;
  const int wave = tid >> 5;
  const int lane = tid & 31;
  const int half = lane >> 4;   // 0: lanes 0-15, 1: lanes 16-31
  const int m16  = lane & 15;
  float* tile = lds + wave * 16 * LDS_STRIDE;

  const float k1 = 1.0f + eps_p[0];
  const int node0 = (blockIdx.x * 8 + wave) * 16;

  // ---- stage h = (1+eps)*x + agg into LDS (16 rows x 64 cols, padded stride) ----
  for (int i = 0; i < 8; ++i) {
    int e4 = lane + i * 32;          // float4 unit index, 0..255
    int r  = e4 >> 4;                // row 0..15
    int c4 = e4 & 15;                // float4 column 0..15
    int node = node0 + r;
    int cn = node < n_nodes ? node : n_nodes - 1;  // clamp (stores are guarded later)
    float4 xv = *((const float4*)(x   + (size_t)cn * 64) + c4);
    float4 av = *((const float4*)(agg + (size_t)cn * 64) + c4);
    float* d = tile + r * LDS_STRIDE + c4 * 4;
    d[0] = k1 * xv.x + av.x;
    d[1] = k1 * xv.y + av.y;
    d[2] = k1 * xv.z + av.z;
    d[3] = k1 * xv.w + av.w;
  }
  __syncthreads();

  // ---- layer 1: acc = h @ W1^T  (A=h 16x64 from LDS, B[k][n]=W1[n][k] from global) ----
  v8f acc[4];
  for (int t = 0; t < 4; ++t)
    for (int j = 0; j < 8; ++j) acc[t][j] = 0.f;

  for (int k = 0; k < 16; ++k) {
    // A fragment: lane<16 holds K=4k+{0,1}, lane>=16 holds K=4k+{2,3}, row M = lane%16
    const float2 af = *(const float2*)(tile + m16 * LDS_STRIDE + 4 * k + 2 * half);
    v2f a; a.x = af.x; a.y = af.y;
    for (int t = 0; t < 4; ++t) {
      int n = t * 16 + m16;
      const float2 bf = *(const float2*)(W1 + (size_t)n * 64 + 4 * k + 2 * half);
      v2f b; b.x = bf.x; b.y = bf.y;
      acc[t] = __builtin_amdgcn_wmma_f32_16x16x4_f32(
          false, a, false, b, (short)0, acc[t], false, false);
    }
  }
  __syncthreads();

  // ---- bias + ReLU; write h1 back to LDS in row-major (C/D layout: lane=N, vgpr=M row) ----
  for (int t = 0; t < 4; ++t) {
    float bv = b1[t * 16 + m16];
    for (int v = 0; v < 8; ++v) {
      float h = acc[t][v] + bv;
      h = h > 0.f ? h : 0.f;
      tile[(v + 8 * half) * LDS_STRIDE + t * 16 + m16] = h;
    }
  }
  __syncthreads();

  // ---- layer 2: acc2 = h1 @ W2^T ----
  v8f acc2[4];
  for (int t = 0; t < 4; ++t)
    for (int j = 0; j < 8; ++j) acc2[t][j] = 0.f;

  for (int k = 0; k < 16; ++k) {
    const float2 af = *(const float2*)(tile + m16 * LDS_STRIDE + 4 * k + 2 * half);
    v2f a; a.x = af.x; a.y = af.y;
    for (int t = 0; t < 4; ++t) {
      int n = t * 16 + m16;
      const float2 bf = *(const float2*)(W2 + (size_t)n * 64 + 4 * k + 2 * half);
      v2f b; b.x = bf.x; b.y = bf.y;
      acc2[t] = __builtin_amdgcn_wmma_f32_16x16x4_f32(
          false, a, false, b, (short)0, acc2[t], false, false);
    }
  }

  // ---- bias + guarded store ----
  for (int t = 0; t < 4; ++t) {
    float bv = b2[t * 16 + m16];
    for (int v = 0; v < 8; ++v) {
      int node = node0 + v + 8 * half;
      if (node < n_nodes)
        out[(size_t)node * 64 + t * 16 + m16] = acc2[t][v] + bv;
    }
  }
}

extern "C" void kernel_launch(void* const* d_in, const int* in_sizes, int n_in,
                              void* d_out, int out_size, void* d_ws, size_t ws_size,
                              hipStream_t stream) {
  const float* x   = (const float*)d_in[0];
  const int*   ei  = (const int*)  d_in[1];
  const float* eps = (const float*)d_in[2];
  const float* W1  = (const float*)d_in[3];
  const float* b1  = (const float*)d_in[4];
  const float* W2  = (const float*)d_in[5];
  const float* b2  = (const float*)d_in[6];
  float* out = (float*)d_out;

  const int n_nodes = in_sizes[0] / 64;
  const int n_edges = in_sizes[1] / 2;
  float* agg = (float*)d_ws;   // n_nodes * 64 floats (~25.6 MB)

  // 1) zero agg
  int n4 = (n_nodes * 64) / 4;
  gin_zero_kernel<<<(n4 + 255) / 256, 256, 0, stream>>>((float4*)agg, n4);

  // 2) scatter-add: one wave per edge
  long long sthreads = (long long)n_edges * 32;
  int sblocks = (int)((sthreads + 255) / 256);
  gin_scatter_kernel<<<sblocks, 256, 0, stream>>>(x, ei, agg, n_edges);

  // 3) fused MLP with fp32 WMMA: 8 tiles (waves) of 16 nodes per block
  int ntiles = (n_nodes + 15) / 16;
  int mblocks = (ntiles + 7) / 8;
  gin_mlp_kernel<<<mblocks, 256, 0, stream>>>(x, agg, eps, W1, b1, W2, b2, out, n_nodes);
}